// KNNModel_20280835572115
// MI455X (gfx1250) — compile-verified
//
#include <hip/hip_runtime.h>

// KNN for MI455X (gfx1250, wave32).
// Kernel 1: d_sq[n] = ||data[n]||^2                    (wave per row)
// Kernel 2: per 16-query tile, scan all N with f32 WMMA GEMM tiles,
//           fused threshold-filtered top-16, then gather rows to out.

typedef float v2f __attribute__((ext_vector_type(2)));
typedef float v8f __attribute__((ext_vector_type(8)));

#define D        256
#define QT       16          // queries per block
#define CAP      256         // candidate buffer per query == section width in columns
#define QS_PAD   264         // LDS row stride (floats) to avoid bank conflicts

// ---------------------------------------------------------------------------
// Kernel 1: squared row norms of the database.
// ---------------------------------------------------------------------------
__global__ __launch_bounds__(256) void knn_dsq_kernel(const float* __restrict__ data,
                                                      float* __restrict__ dsq, int N) {
    int row  = blockIdx.x * 8 + (threadIdx.x >> 5);
    int lane = threadIdx.x & 31;
    if (row >= N) return;
    const float* p = data + (size_t)row * D;
    float s = 0.0f;
    #pragma unroll
    for (int j = 0; j < D / 32; ++j) {
        float v = p[lane + 32 * j];
        s = fmaf(v, v, s);
    }
    #pragma unroll
    for (int off = 16; off > 0; off >>= 1) s += __shfl_xor(s, off, 32);
    if (lane == 0) dsq[row] = s;
}

// ---------------------------------------------------------------------------
// Kernel 2: fused WMMA distance GEMM + filtered top-16 + gather.
// One block = 16 queries, 8 waves. Each wave computes 16x16 score tiles with
// V_WMMA_F32_16X16X4_F32 (K-loop over D=256 in steps of 4).
// ---------------------------------------------------------------------------
__global__ __launch_bounds__(256) void knn_topk_kernel(const float* __restrict__ data,
                                                       const float* __restrict__ query,
                                                       const float* __restrict__ dsq,
                                                       float* __restrict__ out, int N) {
    __shared__ float qs[QT][QS_PAD];       // staged query tile
    __shared__ float bufD[QT][CAP];        // candidate distances
    __shared__ int   bufI[QT][CAP];        // candidate indices
    __shared__ float topD[QT][16];
    __shared__ int   topI[QT][16];
    __shared__ float thr[QT];              // current 16th-best distance
    __shared__ int   wIdx[QT];             // index paired with thr (tie-break)
    __shared__ int   wPos[QT];             // position of worst entry in top16
    __shared__ int   cnt[QT];

    const int tid  = threadIdx.x;
    const int lane = tid & 31;
    const int wave = tid >> 5;
    const int q0   = blockIdx.x * QT;

    // Stage 16 query rows into LDS (fully coalesced).
    for (int i = tid; i < QT * D; i += 256) {
        int r = i >> 8, c = i & (D - 1);
        qs[r][c] = query[(size_t)(q0 + r) * D + c];
    }
    if (tid < QT) {
        thr[tid]  = __builtin_inff();
        wIdx[tid] = 0x7fffffff;
        wPos[tid] = 0;
        cnt[tid]  = 0;
        for (int k = 0; k < 16; ++k) { topD[tid][k] = __builtin_inff(); topI[tid][k] = 0x7fffffff; }
    }
    __syncthreads();

    // WMMA fragment lane mapping (ISA 7.12.2, 32-bit A 16x4 / B 4x16 / C 16x16):
    const int row16 = lane & 15;   // A: M row / B: N column / C: N column
    const int hi    = lane >> 4;   // selects K pair (A/B) and M upper half (C)

    const int numTiles    = N >> 4;           // 3125 full 16-column tiles (N % 16 == 0)
    const int tilesPerSec = CAP >> 4;         // 16 tiles = 256 columns per section
    const int numSec      = (numTiles + tilesPerSec - 1) / tilesPerSec;

    const float* arow = &qs[row16][2 * hi];   // A fragment base (LDS)

    for (int sec = 0; sec < numSec; ++sec) {
        int t0   = sec * tilesPerSec;
        int tEnd = t0 + tilesPerSec;
        if (tEnd > numTiles) tEnd = numTiles;

        for (int t = t0 + wave; t < tEnd; t += 8) {   // uniform per wave -> EXEC all 1s
            const int n0 = t << 4;
            const float* brow = data + (size_t)(n0 + row16) * D + 2 * hi;
            // Hint the next tile this wave will touch into cache.
            __builtin_prefetch(brow + 8 * 16 * D, 0, 1);

            v8f c = {};
            #pragma unroll
            for (int kk = 0; kk < D; kk += 4) {
                v2f a = *(const v2f*)(arow + kk);     // A[M=row16][kk+2*hi .. +1]
                v2f b = *(const v2f*)(brow + kk);     // B[kk+2*hi..][N=row16] = data[n][k]
                c = __builtin_amdgcn_wmma_f32_16x16x4_f32(
                        false, a, false, b, (short)0, c, false, false);
            }

            const int   n  = n0 + row16;
            const float dn = dsq[n];
            #pragma unroll
            for (int i = 0; i < 8; ++i) {
                const int   qi    = i + 8 * hi;                // query row of this accum reg
                const float score = fmaf(-2.0f, c[i], dn);     // ||d||^2 - 2 q.d (rank-equiv)
                if (score < thr[qi]) {
                    int pos = atomicAdd(&cnt[qi], 1);
                    if (pos < CAP) { bufD[qi][pos] = score; bufI[qi][pos] = n; }
                }
            }
        }
        __syncthreads();

        // Merge candidates into running top-16 (one thread per query).
        if (tid < QT) {
            int m = cnt[tid]; if (m > CAP) m = CAP;
            float wd = thr[tid];
            int   wi = wIdx[tid];
            int   wp = wPos[tid];
            for (int i = 0; i < m; ++i) {
                float d  = bufD[tid][i];
                int   ix = bufI[tid][i];
                if (d < wd || (d == wd && ix < wi)) {
                    topD[tid][wp] = d; topI[tid][wp] = ix;
                    // rescan for the new worst (strict (d, idx) total order)
                    wd = topD[tid][0]; wi = topI[tid][0]; wp = 0;
                    for (int k = 1; k < 16; ++k) {
                        float dk = topD[tid][k]; int ik = topI[tid][k];
                        if (dk > wd || (dk == wd && ik > wi)) { wd = dk; wi = ik; wp = k; }
                    }
                }
            }
            thr[tid] = wd; wIdx[tid] = wi; wPos[tid] = wp; cnt[tid] = 0;
        }
        __syncthreads();
    }

    // Sort each query's top-16 ascending by (distance, index) — matches lax.top_k order.
    if (tid < QT) {
        for (int i = 0; i < 15; ++i) {
            int best = i;
            for (int j = i + 1; j < 16; ++j) {
                if (topD[tid][j] < topD[tid][best] ||
                    (topD[tid][j] == topD[tid][best] && topI[tid][j] < topI[tid][best]))
                    best = j;
            }
            float td = topD[tid][i]; topD[tid][i] = topD[tid][best]; topD[tid][best] = td;
            int   ti = topI[tid][i]; topI[tid][i] = topI[tid][best]; topI[tid][best] = ti;
        }
    }
    __syncthreads();

    // Gather: out[q][k][:] = data[topI[q][k]][:]  (16*16*256 floats = 16384 float4)
    for (int u = tid; u < QT * 16 * (D / 4); u += 256) {
        int q   = u >> 10;           // (16 * D/4) = 1024 float4 per query
        int rem = u & 1023;
        int k   = rem >> 6;          // D/4 = 64 float4 per neighbor row
        int d4  = rem & 63;
        float4 v = *(const float4*)(data + (size_t)topI[q][k] * D + d4 * 4);
        *(float4*)(out + ((size_t)(q0 + q) * 16 + k) * D + d4 * 4) = v;
    }
}

extern "C" void kernel_launch(void* const* d_in, const int* in_sizes, int n_in,
                              void* d_out, int out_size, void* d_ws, size_t ws_size,
                              hipStream_t stream) {
    const float* data  = (const float*)d_in[0];
    const float* query = (const float*)d_in[1];
    float*       out   = (float*)d_out;
    float*       dsq   = (float*)d_ws;          // N floats = 200 KB scratch

    const int N = in_sizes[0] / D;              // 50000
    const int Q = in_sizes[1] / D;              // 2048

    knn_dsq_kernel<<<(N + 7) / 8, 256, 0, stream>>>(data, dsq, N);
    knn_topk_kernel<<<Q / QT, 256, 0, stream>>>(data, query, dsq, out, N);
}